// GNOEncoder_8684423873089
// MI455X (gfx1250) — compile-verified
//
#include <hip/hip_runtime.h>
#include <hip/hip_bf16.h>

#define N_PHYS    400000
#define N_LATENT  32768
#define N_EDGES   2097152
#define LIFT      16
#define NTILES    (N_EDGES / 16)   // 131072
#define WSTR      66               // padded row stride (floats) for LDS matrices
#define BLK       128              // 4 waves per block
#define GRID      4096             // 16384 waves total -> exactly 8 tiles per wave

typedef __attribute__((ext_vector_type(2))) float v2f;
typedef __attribute__((ext_vector_type(8))) float v8f;

__device__ __forceinline__ v8f wmma4(v2f a, v2f b, v8f c) {
    // D(16x16) = A(16x4) * B(4x16) + C   -- f32 tensor path on gfx1250
    return __builtin_amdgcn_wmma_f32_16x16x4_f32(
        /*neg_a=*/false, a, /*neg_b=*/false, b,
        /*c_mod=*/(short)0, c, /*reuse_a=*/false, /*reuse_b=*/false);
}

__device__ __forceinline__ float fast_tanh(float x) {
#if __has_builtin(__builtin_amdgcn_tanhf)
    return __builtin_amdgcn_tanhf(x);        // v_tanh_f32 (gfx1250 TRANS op)
#elif __has_builtin(__builtin_amdgcn_tanh_f32)
    return __builtin_amdgcn_tanh_f32(x);
#else
    return tanhf(x);
#endif
}

__device__ __forceinline__ float gelu_tanh(float x) {
    const float k0 = 0.7978845608028654f;  // sqrt(2/pi)
    const float k1 = 0.044715f;
    float x3 = x * x * x;
    return 0.5f * x * (1.0f + fast_tanh(k0 * fmaf(k1, x3, x)));
}

// One dense 64->64 layer for a 16-edge tile: stage (16 x 64, stride WSTR) -> stage
__device__ __forceinline__ void layer64(float* stage, const float* sW,
                                        const float* sBias, int row, int half,
                                        bool act) {
    // Load all A chunks (K = 64 -> 16 chunks of 4) into registers first.
    v2f a[16];
    const float* arow = stage + row * WSTR + half * 2;
#pragma unroll
    for (int kc = 0; kc < 16; ++kc) a[kc] = *(const v2f*)(arow + kc * 4);

#pragma unroll
    for (int nt = 0; nt < 4; ++nt) {
        v8f c = {0.f, 0.f, 0.f, 0.f, 0.f, 0.f, 0.f, 0.f};
        const float* brow = sW + (nt * 16 + row) * WSTR + half * 2;
#pragma unroll
        for (int kc = 0; kc < 16; ++kc) {
            v2f b = *(const v2f*)(brow + kc * 4);
            c = wmma4(a[kc], b, c);
        }
        float bias = sBias[nt * 16 + row];
#pragma unroll
        for (int i = 0; i < 8; ++i) {
            float x = c[i] + bias;
            c[i] = act ? gelu_tanh(x) : x;
        }
        // C layout: VGPR i -> edge = i + 8*half, feat = nt*16 + row
#pragma unroll
        for (int i = 0; i < 8; ++i)
            stage[(i + 8 * half) * WSTR + nt * 16 + row] = c[i];
    }
}

__global__ __launch_bounds__(BLK) void gno_encoder_main(
    const float* __restrict__ phys, const float* __restrict__ lat,
    const int* __restrict__ etgt, const int* __restrict__ esrc,
    const float* __restrict__ gWl, const float* __restrict__ gbl,
    const float* __restrict__ gW0, const float* __restrict__ gb0,
    const float* __restrict__ gW1, const float* __restrict__ gb1,
    const float* __restrict__ gW2, const float* __restrict__ gb2,
    const float* __restrict__ gW3, const float* __restrict__ gb3,
    float* __restrict__ ws_sum, float* __restrict__ ws_cnt) {
    // ---- LDS: weights (padded), biases, per-wave staging + edge metadata ----
    __shared__ float sW0[64 * 8];        // K padded 6->8, stride 8
    __shared__ float sW1[64 * WSTR];
    __shared__ float sW2[64 * WSTR];
    __shared__ float sW3[16 * WSTR];
    __shared__ float sB0[64], sB1[64], sB2[64], sB3[16];
    __shared__ float sWl[16][4];
    __shared__ float sBl[16];
    __shared__ float sStage[4 * 16 * WSTR];   // one 16x64(+pad) tile per wave
    __shared__ int   sMetaT[4 * 16];          // tgt index per edge
    __shared__ float sMetaP[4 * 16 * 4];      // phys coords per edge (+pad)

    const int tid = threadIdx.x;
    // cooperative weight preload
    for (int i = tid; i < 64 * 8; i += BLK) {
        int r = i >> 3, k = i & 7;
        sW0[i] = (k < 6) ? gW0[r * 6 + k] : 0.0f;
    }
    for (int i = tid; i < 64 * 64; i += BLK) {
        int r = i >> 6, k = i & 63;
        sW1[r * WSTR + k] = gW1[i];
        sW2[r * WSTR + k] = gW2[i];
    }
    for (int i = tid; i < 16 * 64; i += BLK) {
        int r = i >> 6, k = i & 63;
        sW3[r * WSTR + k] = gW3[i];
    }
    for (int i = tid; i < 64; i += BLK) { sB0[i] = gb0[i]; sB1[i] = gb1[i]; sB2[i] = gb2[i]; }
    if (tid < 16) {
        sB3[tid] = gb3[tid];
        sBl[tid] = gbl[tid];
        sWl[tid][0] = gWl[tid * 3 + 0];
        sWl[tid][1] = gWl[tid * 3 + 1];
        sWl[tid][2] = gWl[tid * 3 + 2];
        sWl[tid][3] = 0.0f;
    }
    __syncthreads();

    const int lane  = tid & 31;
    const int row   = lane & 15;     // M row / N col within tile
    const int half  = lane >> 4;     // 0: K 0-1, 1: K 2-3 of each chunk
    const int wslot = tid >> 5;
    float* stage = sStage + wslot * (16 * WSTR);

    const int gwave = blockIdx.x * (BLK >> 5) + wslot;
    const int nw    = GRID * (BLK >> 5);

    for (int tile = gwave; tile < NTILES; tile += nw) {
        const int e = (tile << 4) + row;
        if (tile + nw < NTILES) {  // uniform prefetch of next tile's indices
            __builtin_prefetch(esrc + e + (nw << 4), 0, 0);
            __builtin_prefetch(etgt + e + (nw << 4), 0, 0);
        }
        const int src = esrc[e];
        const int tgt = etgt[e];

        // Gather k_in = [px,py,pz,lx,ly,lz] directly into WMMA A layout (K pad 8)
        float c0 = phys[src * 3 + (half ? 2 : 0)];                 // px | pz
        float c1 = half ? lat[tgt * 3 + 0] : phys[src * 3 + 1];    // py | lx
        float c2 = half ? 0.0f : lat[tgt * 3 + 1];                 // ly | 0
        float c3 = half ? 0.0f : lat[tgt * 3 + 2];                 // lz | 0
        v2f a00 = {c0, c1};   // K chunk 0 (k 0..3)
        v2f a01 = {c2, c3};   // K chunk 1 (k 4..7, zero padded)

        // edge metadata for the epilogue (uniform stores, no divergence)
        sMetaT[wslot * 16 + row] = tgt;
        sMetaP[(wslot * 16 + row) * 4 + half * 2]     = c0;  // px / pz
        sMetaP[(wslot * 16 + row) * 4 + 1 + half * 2] = c1;  // py / (pad)
        if (!half) atomicAdd(ws_cnt + tgt, 1.0f);

        // ---- layer 0: K=8 (padded), N=64 ----
#pragma unroll
        for (int nt = 0; nt < 4; ++nt) {
            v8f c = {0.f, 0.f, 0.f, 0.f, 0.f, 0.f, 0.f, 0.f};
            const float* brow = sW0 + (nt * 16 + row) * 8 + half * 2;
            c = wmma4(a00, *(const v2f*)(brow + 0), c);
            c = wmma4(a01, *(const v2f*)(brow + 4), c);
            float bias = sB0[nt * 16 + row];
#pragma unroll
            for (int i = 0; i < 8; ++i) c[i] = gelu_tanh(c[i] + bias);
#pragma unroll
            for (int i = 0; i < 8; ++i)
                stage[(i + 8 * half) * WSTR + nt * 16 + row] = c[i];
        }

        // ---- layers 1,2: K=64, N=64, gelu ----
        layer64(stage, sW1, sB1, row, half, true);
        layer64(stage, sW2, sB2, row, half, true);

        // ---- layer 3: K=64, N=16, no activation; keep D in registers ----
        v2f a[16];
        const float* arow = stage + row * WSTR + half * 2;
#pragma unroll
        for (int kc = 0; kc < 16; ++kc) a[kc] = *(const v2f*)(arow + kc * 4);
        v8f d = {0.f, 0.f, 0.f, 0.f, 0.f, 0.f, 0.f, 0.f};
        {
            const float* brow = sW3 + row * WSTR + half * 2;
#pragma unroll
            for (int kc = 0; kc < 16; ++kc)
                d = wmma4(a[kc], *(const v2f*)(brow + kc * 4), d);
        }
        const float bias3 = sB3[row];

        // ---- epilogue: msg = k * lifted(src);  atomic segment-sum over tgt ----
        const float wl0 = sWl[row][0], wl1 = sWl[row][1], wl2 = sWl[row][2];
        const float blf = sBl[row];
#pragma unroll
        for (int i = 0; i < 8; ++i) {
            int edge = i + 8 * half;            // D layout: VGPR i -> edge, feat = row
            int m = wslot * 16 + edge;
            int t = sMetaT[m];
            float px = sMetaP[m * 4 + 0];
            float py = sMetaP[m * 4 + 1];
            float pz = sMetaP[m * 4 + 2];
            float lift = fmaf(px, wl0, fmaf(py, wl1, fmaf(pz, wl2, blf)));
            atomicAdd(ws_sum + t * LIFT + row, (d[i] + bias3) * lift);
        }
    }
}

__global__ __launch_bounds__(256) void gno_finalize(const float* __restrict__ sum,
                                                    const float* __restrict__ cnt,
                                                    float* __restrict__ out) {
    int i = blockIdx.x * 256 + threadIdx.x;
    if (i < N_LATENT * LIFT) out[i] = sum[i] / fmaxf(cnt[i >> 4], 1.0f);
}

extern "C" void kernel_launch(void* const* d_in, const int* in_sizes, int n_in,
                              void* d_out, int out_size, void* d_ws, size_t ws_size,
                              hipStream_t stream) {
    const float* phys = (const float*)d_in[0];
    const float* lat  = (const float*)d_in[1];
    const int*   etgt = (const int*)d_in[2];
    const int*   esrc = (const int*)d_in[3];
    const float* Wl = (const float*)d_in[4];
    const float* bl = (const float*)d_in[5];
    const float* W0 = (const float*)d_in[6];
    const float* b0 = (const float*)d_in[7];
    const float* W1 = (const float*)d_in[8];
    const float* b1 = (const float*)d_in[9];
    const float* W2 = (const float*)d_in[10];
    const float* b2 = (const float*)d_in[11];
    const float* W3 = (const float*)d_in[12];
    const float* b3 = (const float*)d_in[13];

    float* ws_sum = (float*)d_ws;                      // N_LATENT*LIFT floats
    float* ws_cnt = ws_sum + (size_t)N_LATENT * LIFT;  // N_LATENT floats

    hipMemsetAsync(d_ws, 0, ((size_t)N_LATENT * LIFT + N_LATENT) * sizeof(float), stream);

    gno_encoder_main<<<GRID, BLK, 0, stream>>>(
        phys, lat, etgt, esrc, Wl, bl, W0, b0, W1, b1, W2, b2, W3, b3,
        ws_sum, ws_cnt);

    int n = N_LATENT * LIFT;
    gno_finalize<<<(n + 255) / 256, 256, 0, stream>>>(ws_sum, ws_cnt, (float*)d_out);
}